// HashEncoding_88837103551034
// MI455X (gfx1250) — compile-verified
//
#include <hip/hip_runtime.h>
#include <cmath>

// ---------------------------------------------------------------------------
// Instant-NGP hash-grid encoding for MI455X (gfx1250).
//   N points x 16 levels x 8 corners, T = 2^19 slots/level, 2 feats/level.
//
// Roofline: 67M random 8B gathers from a 67MB table (resident in 192MB L2)
// dominate. Streamed I/O (points in, 67MB encodings out) uses CDNA5
// non-temporal hints so it never evicts the table. Output is staged through
// LDS and streamed with the gfx1250 async LDS->memory engine
// (global_store_async_from_lds_b128, ASYNCcnt) for fully-coalesced NT writes
// with no VGPR round trip.
// ---------------------------------------------------------------------------

#define NUM_LEVELS 16
#define LOG2_T     19
#define T_SIZE     (1u << LOG2_T)
#define T_MASK     (T_SIZE - 1u)
#define PRIME_Y    2654435761u
#define PRIME_Z    805459861u

#define BLOCK      256
#define ROW_BYTES  136u   // 32 floats (128B) + 8B pad: t*34 dwords mod 64 is a
                          // permutation -> conflict-free ds_store_b64

typedef float v2f __attribute__((ext_vector_type(2)));

struct Scalings { float s[NUM_LEVELS]; };

__global__ __launch_bounds__(BLOCK) void HashEncoding_88837103551034_kernel(
    const float* __restrict__ x,          // [N,3]
    const float* __restrict__ table,      // [T*NUM_LEVELS, 2]
    float* __restrict__ out,              // [N, NUM_LEVELS*2]
    int n,
    Scalings sc)
{
    __shared__ unsigned char smem[BLOCK * ROW_BYTES];   // ~34 KB staging

    const int t = threadIdx.x;
    const int i = blockIdx.x * BLOCK + t;
    const bool full = ((blockIdx.x + 1) * BLOCK <= n);  // whole block valid?

    if (i < n) {
        // Point coords: streamed once -> non-temporal load (keep L2 for table).
        const float px = __builtin_nontemporal_load(x + 3 * (size_t)i + 0);
        const float py = __builtin_nontemporal_load(x + 3 * (size_t)i + 1);
        const float pz = __builtin_nontemporal_load(x + 3 * (size_t)i + 2);

        // map [-1,1] -> [0,1] in fp32, same as reference
        const float x0 = px * 0.5f + 0.5f;
        const float y0 = py * 0.5f + 0.5f;
        const float z0 = pz * 0.5f + 0.5f;

        v2f* const myrow = (v2f*)(smem + (unsigned)t * ROW_BYTES);
        float* const outp = out + (size_t)i * (NUM_LEVELS * 2);

#pragma unroll
        for (int l = 0; l < NUM_LEVELS; ++l) {
            const float s = sc.s[l];
            const float sx = x0 * s, sy = y0 * s, sz = z0 * s;

            const float fx = floorf(sx), fy = floorf(sy), fz = floorf(sz);
            const float ox = sx - fx,   oy = sy - fy,   oz = sz - fz;
            const float wx0 = 1.0f - ox, wy0 = 1.0f - oy, wz0 = 1.0f - oz;

            // Reference hash is int64, but corners >= 0 and T = 2^19 is a
            // power of two, so the masked XOR-hash is bit-exact in uint32.
            const unsigned hx0 = (unsigned)(int)fx;
            const unsigned hx1 = (unsigned)(int)ceilf(sx);
            const unsigned hy0 = (unsigned)(int)fy        * PRIME_Y;
            const unsigned hy1 = (unsigned)(int)ceilf(sy) * PRIME_Y;
            const unsigned hz0 = (unsigned)(int)fz        * PRIME_Z;
            const unsigned hz1 = (unsigned)(int)ceilf(sz) * PRIME_Z;

            // Per-level table base is uniform -> SGPR base + VGPR offset.
            const v2f* __restrict__ tl = (const v2f*)table + (size_t)l * T_SIZE;

            v2f acc = {0.0f, 0.0f};
#pragma unroll
            for (int b = 0; b < 8; ++b) {
                // CORNER_BITS: bit2 -> dim x, bit1 -> dim y, bit0 -> dim z
                const unsigned hx = (b & 4) ? hx1 : hx0;
                const unsigned hy = (b & 2) ? hy1 : hy0;
                const unsigned hz = (b & 1) ? hz1 : hz0;
                const unsigned h  = (hx ^ hy ^ hz) & T_MASK;

                const float wx = (b & 4) ? ox : wx0;
                const float wy = (b & 2) ? oy : wy0;
                const float wz = (b & 1) ? oz : wz0;
                const float w  = wx * wy * wz;   // same order as jnp.prod

                const v2f f = tl[h];             // 8B gather (RT: stay in L2)
                acc.x += w * f.x;
                acc.y += w * f.y;
            }

            if (full) {
                myrow[l] = acc;                  // conflict-free ds_store_b64
            } else {
                __builtin_nontemporal_store(acc, (v2f*)(outp + 2 * l));
            }
        }
    }

    if (full) {
        __syncthreads();   // s_wait_dscnt 0 + barrier: LDS rows complete

        // Stream the block's contiguous 32KB output region with the async
        // LDS->memory engine: fully coalesced (512B per wave instruction),
        // non-temporal, no VGPR round trip. S_ENDPGM's implicit wait-idle
        // covers ASYNCcnt completion.
        const unsigned lbase =
            (unsigned)(unsigned long long)(const void*)smem;  // LDS byte offset
        const unsigned long long gbase =
            (unsigned long long)(out + (size_t)blockIdx.x * (BLOCK * NUM_LEVELS * 2));

#pragma unroll
        for (int j = 0; j < 8; ++j) {
            const unsigned k   = (unsigned)(j * BLOCK + t);   // float4 index
            const unsigned row = k >> 3;                      // 8 float4 / row
            const unsigned col = k & 7u;
            const unsigned lds_off = lbase + row * ROW_BYTES + col * 16u;
            const unsigned long long gaddr = gbase + (unsigned long long)k * 16u;
            asm volatile(
                "global_store_async_from_lds_b128 %0, %1, off th:TH_STORE_NT"
                : : "v"(gaddr), "v"(lds_off) : "memory");
        }
    }
}

extern "C" void kernel_launch(void* const* d_in, const int* in_sizes, int n_in,
                              void* d_out, int out_size, void* d_ws, size_t ws_size,
                              hipStream_t stream) {
    (void)n_in; (void)out_size; (void)d_ws; (void)ws_size;

    const float* x     = (const float*)d_in[0];   // [N,3] fp32
    const float* table = (const float*)d_in[1];   // [T*16, 2] fp32
    float*       out   = (float*)d_out;           // [N, 32] fp32

    const int n = in_sizes[0] / 3;

    // Per-level resolutions computed on the HOST in double precision with the
    // exact libm formula the reference uses (floor(16 * growth^l) is
    // borderline at l = 5,10,15, so hardcoding risks off-by-one).
    Scalings sc;
    const double growth = exp((log(1024.0) - log(16.0)) / 15.0);
    for (int l = 0; l < NUM_LEVELS; ++l)
        sc.s[l] = (float)floor(16.0 * pow(growth, (double)l));

    const int blocks = (n + BLOCK - 1) / BLOCK;
    HashEncoding_88837103551034_kernel<<<blocks, BLOCK, 0, stream>>>(
        x, table, out, n, sc);
}